// LossTotal_32521492366089
// MI455X (gfx1250) — compile-verified
//
#include <hip/hip_runtime.h>
#include <math.h>

// Problem constants from the reference
#define HDIM 704
#define WDIM 704
#define HW   (704 * 704)
#define MREF 32
#define PPOS 128
#define NNEG 129
#define BLAST 3   // only losses[-1] is returned

typedef float v2f __attribute__((ext_vector_type(2)));
typedef float v8f __attribute__((ext_vector_type(8)));

__device__ __forceinline__ float softplus_f(float d) {
    // log(1 + exp(d)), numerically stable; equals -log_softmax for 2-class CE
    return fmaxf(d, 0.0f) + log1pf(expf(-fabsf(d)));
}

__device__ __forceinline__ float smooth_l1_f(float d) {
    float a = fabsf(d);
    return (a < 1.0f) ? (0.5f * d * d) : (a - 0.5f);
}

__global__ __launch_bounds__(512)
void loss_total_kernel(const float* __restrict__ ref_boxes,    // (B,32,7)
                       const float* __restrict__ pred_class,   // (B,4,H,W)
                       const float* __restrict__ pred_regress, // (B,14,H,W)
                       const float* __restrict__ anchor,       // (2,7,H,W)
                       const int*   __restrict__ pos_idx,      // (B,128,2)
                       const int*   __restrict__ neg_idx,      // (B,129,2)
                       const int*   __restrict__ reg_idx,      // (B,32,1,2)
                       float*       __restrict__ out)          // scalar
{
    const int tid = threadIdx.x;

    const float* pc   = pred_class   + (size_t)BLAST * 4  * HW;
    const float* pr   = pred_regress + (size_t)BLAST * 14 * HW;
    const float* rb   = ref_boxes    + BLAST * MREF * 7;
    const int*   pidx = pos_idx      + BLAST * PPOS * 2;
    const int*   nidx = neg_idx      + BLAST * NNEG * 2;
    const int*   ridx = reg_idx      + BLAST * MREF * 2;   // R == 1

    float contrib = 0.0f;

    if (tid < PPOS) {
        // ---- positive classification points: label 1 on both anchor pairs
        const int y = pidx[2 * tid], x = pidx[2 * tid + 1];
        const size_t off = (size_t)y * WDIM + (size_t)x;
        const float l0 = pc[off];
        const float l1 = pc[(size_t)1 * HW + off];
        const float l2 = pc[(size_t)2 * HW + off];
        const float l3 = pc[(size_t)3 * HW + off];
        contrib = (softplus_f(l0 - l1) + softplus_f(l2 - l3)) * (1.0f / PPOS);
    } else if (tid < PPOS + NNEG) {
        // ---- negative classification points: label 0 on both anchor pairs
        const int q = tid - PPOS;
        const int y = nidx[2 * q], x = nidx[2 * q + 1];
        const size_t off = (size_t)y * WDIM + (size_t)x;
        const float l0 = pc[off];
        const float l1 = pc[(size_t)1 * HW + off];
        const float l2 = pc[(size_t)2 * HW + off];
        const float l3 = pc[(size_t)3 * HW + off];
        contrib = (softplus_f(l1 - l0) + softplus_f(l3 - l2)) * (1.0f / NNEG);
    } else if (tid < PPOS + NNEG + MREF) {
        // ---- regression: one thread per reference box (R=1, 2 anchors, 7 ch)
        const int m = tid - (PPOS + NNEG);
        const int y = ridx[2 * m], x = ridx[2 * m + 1];
        const size_t off = (size_t)y * WDIM + (size_t)x;

        float refv[7];
        #pragma unroll
        for (int c = 0; c < 7; ++c) refv[c] = rb[m * 7 + c];

        float s = 0.0f;
        #pragma unroll
        for (int a = 0; a < 2; ++a) {
            float ab[7], pb[7];
            #pragma unroll
            for (int c = 0; c < 7; ++c) {
                ab[c] = anchor[(size_t)(a * 7 + c) * HW + off];
                pb[c] = pr[(size_t)(a * 7 + c) * HW + off];
            }
            const float diag = sqrtf(ab[3] * ab[3] + ab[4] * ab[4]);
            float offv[7];
            offv[0] = (refv[0] - ab[0]) / diag;
            offv[1] = (refv[1] - ab[1]) / diag;
            offv[2] = (refv[2] - ab[2]) / ab[5];
            offv[3] = logf(refv[3] / ab[3]);
            offv[4] = logf(refv[4] / ab[4]);
            offv[5] = logf(refv[5] / ab[5]);
            const float dth = refv[6] - ab[6];
            offv[6] = atan2f(sinf(dth), cosf(dth));
            #pragma unroll
            for (int c = 0; c < 7; ++c) s += smooth_l1_f(pb[c] - offv[c]);
        }
        // reg mean over (R,2,7)=14 elements, summed over m, scaled by GAIN=2
        contrib = 2.0f * s * (1.0f / 14.0f);
    }

    // ---- wave32 reduction within each of the 16 waves
    #pragma unroll
    for (int o = 16; o > 0; o >>= 1) contrib += __shfl_down(contrib, o, 32);

    __shared__ float warp_sum[16];
    const int lane = tid & 31;
    const int wid  = tid >> 5;
    if (lane == 0) warp_sum[wid] = contrib;
    __syncthreads();

    // ---- final cross-wave reduction via V_WMMA_F32_16X16X4_F32 (exact f32)
    // Wave-uniform branch: all 32 lanes of wave 0 active -> EXEC all ones.
    if (tid < 32) {
        const float p = (lane < 16) ? warp_sum[lane] : 0.0f;

        // A (16x4 f32): lanes 0..15 VGPR0 = A[M=lane, K=0] = partial[lane];
        // all other K slots zero. B = all ones => D[m,n] = partial[m].
        v2f a;
        a.x = (lane < 16) ? p : 0.0f;  // K=0 (lanes 0-15) / K=2 (lanes 16-31)
        a.y = 0.0f;                    // K=1 / K=3
        v2f bones;
        bones.x = 1.0f;
        bones.y = 1.0f;
        v8f cacc = {};

        v8f d = __builtin_amdgcn_wmma_f32_16x16x4_f32(
            /*neg_a=*/false, a, /*neg_b=*/false, bones,
            /*c_mod=*/(short)0, cacc, /*reuse_a=*/false, /*reuse_b=*/false);

        // Lane n holds D[0..7, n] (rows 0-7); lane n+16 holds rows 8-15.
        float r = d[0] + d[1] + d[2] + d[3] + d[4] + d[5] + d[6] + d[7];
        r += __shfl_xor(r, 16, 32);   // combine the two row halves

        if (lane == 0) out[0] = r;
    }
}

extern "C" void kernel_launch(void* const* d_in, const int* in_sizes, int n_in,
                              void* d_out, int out_size, void* d_ws, size_t ws_size,
                              hipStream_t stream) {
    (void)in_sizes; (void)n_in; (void)out_size; (void)d_ws; (void)ws_size;

    const float* ref_boxes    = (const float*)d_in[0];
    const float* pred_class   = (const float*)d_in[1];
    const float* pred_regress = (const float*)d_in[2];
    const float* anchor       = (const float*)d_in[3];
    const int*   pos_idx      = (const int*)d_in[4];
    const int*   neg_idx      = (const int*)d_in[5];
    const int*   reg_idx      = (const int*)d_in[6];
    float*       out          = (float*)d_out;

    // All useful work (289 gather points + final reduce) fits in one WGP:
    // the kernel is latency-bound on ~2K random cache lines, so a single
    // 512-thread (16-wave) block is the minimum-latency configuration.
    loss_total_kernel<<<dim3(1), dim3(512), 0, stream>>>(
        ref_boxes, pred_class, pred_regress, anchor,
        pos_idx, neg_idx, reg_idx, out);
}